// NIBLoss_37486474560077
// MI455X (gfx1250) — compile-verified
//
#include <hip/hip_runtime.h>
#include <math.h>

typedef __attribute__((ext_vector_type(2))) float v2f;
typedef __attribute__((ext_vector_type(8))) float v8f;

#define NROWS 8192
#define DDIM  64
#define NTILE (NROWS / 16)          // 512 16x16 tiles per dimension
#define JCHUNK 8                    // grid.y split of the j-tile range
#define TILES_PER_CHUNK (NTILE / JCHUNK)  // 64
#define WAVES 4                     // waves per block (wave32)

// workspace layout in floats
#define WS_DIST 0                   // [0]      distortion accumulator
#define WS_SQ   8                   // [8..8+N) row squared norms
#define WS_ROW  (8 + NROWS)         // [..+N)   per-row exp sums
#define WS_ZERO_FLOATS (8 + 2 * NROWS)

__global__ void zero_ws_kernel(float* ws, int n) {
    int i = blockIdx.x * blockDim.x + threadIdx.x;
    if (i < n) ws[i] = 0.0f;
}

// sq[r] = sum_d encoded[r][d]^2   (one thread per row, float4 loads)
__global__ void row_sq_kernel(const float* __restrict__ E, float* __restrict__ sq) {
    int row = blockIdx.x * blockDim.x + threadIdx.x;
    if (row >= NROWS) return;
    const float4* p = (const float4*)(E + row * DDIM);
    float s = 0.0f;
#pragma unroll
    for (int i = 0; i < DDIM / 4; ++i) {
        float4 v = p[i];
        s += v.x * v.x + v.y * v.y + v.z * v.z + v.w * v.w;
    }
    sq[row] = s;
}

// sum of (y_pred - y_true)^2 into ws[WS_DIST]
__global__ void mse_kernel(const float* __restrict__ yt, const float* __restrict__ yp,
                           float* __restrict__ acc) {
    __shared__ float red[256];
    int tid = threadIdx.x;
    int idx = blockIdx.x * blockDim.x + tid;
    int stride = gridDim.x * blockDim.x;
    float s = 0.0f;
    const int n4 = NROWS * DDIM / 4;
    for (int i = idx; i < n4; i += stride) {
        float4 a = ((const float4*)yt)[i];
        float4 b = ((const float4*)yp)[i];
        float dx = b.x - a.x, dy = b.y - a.y, dz = b.z - a.z, dw = b.w - a.w;
        s += dx * dx + dy * dy + dz * dz + dw * dw;
    }
    red[tid] = s;
    __syncthreads();
    for (int off = 128; off > 0; off >>= 1) {
        if (tid < off) red[tid] += red[tid + off];
        __syncthreads();
    }
    if (tid == 0) atomicAdd(acc, red[0]);
}

// Main kernel: for i-tile = blockIdx.x, sweep j-tiles of chunk blockIdx.y.
// rowsum[i] += sum_j exp(-0.5 * max(sq[i]+sq[j]-2*G[i][j], 0))
__global__ void __launch_bounds__(WAVES * 32)
kde_tiles_kernel(const float* __restrict__ E, const float* __restrict__ sq,
                 float* __restrict__ rowsum) {
    const int lane = threadIdx.x & 31;
    const int wave = threadIdx.x >> 5;
    const int half = lane >> 4;     // 0: K=0,1 / rows 0-7 of C ; 1: K=2,3 / rows 8-15
    const int l15  = lane & 15;
    const int i0   = blockIdx.x * 16;

    // A tile (16x4 f32 layout): lane holds E[i0+l15][4s+2*half .. +1] for s=0..15.
    v2f a[16];
    const float* arow = E + (size_t)(i0 + l15) * DDIM + 2 * half;
#pragma unroll
    for (int s = 0; s < 16; ++s)
        a[s] = *(const v2f*)(arow + 4 * s);

    // sq of my 8 output rows (M = v + 8*half), invariant over the j loop
    float sqi[8];
#pragma unroll
    for (int v = 0; v < 8; ++v)
        sqi[v] = sq[i0 + 8 * half + v];

    float racc[8];
#pragma unroll
    for (int v = 0; v < 8; ++v) racc[v] = 0.0f;

    const int jt0 = blockIdx.y * TILES_PER_CHUNK;
    for (int t = wave; t < TILES_PER_CHUNK; t += WAVES) {
        const int j0 = (jt0 + t) * 16;
        // B tile (4x16 f32 layout) == A-pattern of rows j0.. because B = E^T block
        const float* brow = E + (size_t)(j0 + l15) * DDIM + 2 * half;
        v2f b[16];
#pragma unroll
        for (int s = 0; s < 16; ++s)
            b[s] = *(const v2f*)(brow + 4 * s);
        const float sqj = sq[j0 + l15];

        v8f c = {};
#pragma unroll
        for (int s = 0; s < 16; ++s)
            c = __builtin_amdgcn_wmma_f32_16x16x4_f32(
                    /*neg_a=*/false, a[s], /*neg_b=*/false, b[s],
                    /*c_mod=*/(short)0, c, /*reuse_a=*/false, /*reuse_b=*/false);

        // lane's C elements: row M = v + 8*half, col N = l15 (within this j-tile)
#pragma unroll
        for (int v = 0; v < 8; ++v) {
            float d2 = sqi[v] + sqj - 2.0f * c[v];
            d2 = fmaxf(d2, 0.0f);
            racc[v] += __expf(-0.5f * d2);
        }
    }

    // reduce over the 16 lanes of each half-wave (columns of the row block)
#pragma unroll
    for (int v = 0; v < 8; ++v) {
        float s = racc[v];
#pragma unroll
        for (int m = 1; m < 16; m <<= 1)
            s += __shfl_xor(s, m, 32);
        racc[v] = s;
    }
    if (l15 == 0) {
#pragma unroll
        for (int v = 0; v < 8; ++v)
            atomicAdd(&rowsum[i0 + 8 * half + v], racc[v]);
    }
}

// Final scalars: kde = mean_i log(rowsum[i]); IXT, distortion, loss.
__global__ void finalize_kernel(const float* __restrict__ rowsum,
                                const float* __restrict__ dist_acc,
                                float* __restrict__ out) {
    __shared__ float red[256];
    int tid = threadIdx.x;
    float s = 0.0f;
    for (int i = tid; i < NROWS; i += 256)
        s += logf(rowsum[i]);
    red[tid] = s;
    __syncthreads();
    for (int off = 128; off > 0; off >>= 1) {
        if (tid < off) red[tid] += red[tid + off];
        __syncthreads();
    }
    if (tid == 0) {
        const float LN2  = 0.69314718055994531f;
        const float LOGN = 9.01091334727928791f;  // ln(8192)
        float kde = red[0] / (float)NROWS;
        float ixt = (LOGN - kde) / LN2;
        float distortion = dist_acc[0] / (float)(NROWS * DDIM);
        float loss = ixt + 500.0f * distortion;
        out[0] = loss;
        out[1] = ixt;
        out[2] = distortion;
    }
}

extern "C" void kernel_launch(void* const* d_in, const int* in_sizes, int n_in,
                              void* d_out, int out_size, void* d_ws, size_t ws_size,
                              hipStream_t stream) {
    const float* y_true  = (const float*)d_in[0];
    const float* y_pred  = (const float*)d_in[1];
    const float* encoded = (const float*)d_in[2];
    float* ws  = (float*)d_ws;
    float* out = (float*)d_out;

    // zero accumulators (kernel, to stay graph-capture safe)
    zero_ws_kernel<<<(WS_ZERO_FLOATS + 255) / 256, 256, 0, stream>>>(ws, WS_ZERO_FLOATS);

    // row squared norms of encoded
    row_sq_kernel<<<NROWS / 256, 256, 0, stream>>>(encoded, ws + WS_SQ);

    // distortion partial sums
    mse_kernel<<<256, 256, 0, stream>>>(y_true, y_pred, ws + WS_DIST);

    // Gram-tile sweep with f32 WMMA + exp accumulation
    dim3 grid(NTILE, JCHUNK);
    kde_tiles_kernel<<<grid, WAVES * 32, 0, stream>>>(encoded, ws + WS_SQ, ws + WS_ROW);

    // final scalars
    finalize_kernel<<<1, 256, 0, stream>>>(ws + WS_ROW, ws + WS_DIST, out);
}